// Net_62251255988834
// MI455X (gfx1250) — compile-verified
//
#include <hip/hip_runtime.h>

// ---------------------------------------------------------------------------
// GCN (3-layer) for MI455X / gfx1250, wave32.
// Edge scatter (L2-atomic bound) dominates; GEMMs use v_wmma_f32_16x16x4_f32.
// ---------------------------------------------------------------------------

typedef float v2f __attribute__((ext_vector_type(2)));
typedef float v4f __attribute__((ext_vector_type(4)));
typedef float v8f __attribute__((ext_vector_type(8)));

static __device__ __forceinline__ void atomAddF(float* p, float v) {
    // Lowers to global_atomic_add_f32 (non-returning: result unused).
    __hip_atomic_fetch_add(p, v, __ATOMIC_RELAXED, __HIP_MEMORY_SCOPE_AGENT);
}

// ------------------------------- utility -----------------------------------

__global__ void k_fill(float* __restrict__ p, float v, long long n) {
    long long i = (long long)blockIdx.x * blockDim.x + threadIdx.x;
    long long stride = (long long)gridDim.x * blockDim.x;
    for (; i < n; i += stride) p[i] = v;
}

__global__ void k_deg(const int* __restrict__ dst, float* __restrict__ deg, int E) {
    int e = blockIdx.x * blockDim.x + threadIdx.x;
    if (e < E) atomAddF(&deg[dst[e]], 1.0f);
}

__global__ void k_rsqrt(float* __restrict__ d, int N) {
    int i = blockIdx.x * blockDim.x + threadIdx.x;
    if (i < N) d[i] = rsqrtf(d[i]);
}

// Build unified edge list (real edges + self loops) with precomputed norm.
__global__ void k_edges(const int* __restrict__ src, const int* __restrict__ dst,
                        const float* __restrict__ dinv,
                        int* __restrict__ s32, int* __restrict__ d32,
                        float* __restrict__ nrm, int E, int N) {
    int e = blockIdx.x * blockDim.x + threadIdx.x;
    if (e >= E + N) return;
    int s, d;
    if (e < E) { s = src[e]; d = dst[e]; }
    else       { s = d = e - E; }
    s32[e] = s;
    d32[e] = d;
    nrm[e] = dinv[s] * dinv[d];
}

// ------------------------------- GEMM (WMMA) -------------------------------
// C[N,H] = act(A)[N,128] @ W[128,H],  act = FUSE ? relu(.+biasIn[k]) : id.
// H is a template parameter so every weight load folds k0*H into the 24-bit
// immediate offset (no per-iteration 64-bit address VALU).
// One wave -> 16x32 output tile (two independent WMMA accumulators).
// Out-of-range rows/cols are CLAMPED on load (not predicated): garbage only
// lands in tile columns/rows that are never stored -> zero divergence.
//
// f32 16x4 A layout  : lane l: row M=l&15, holds K = k0+2*(l>>4)+{0,1}
// f32 4x16 B layout  : lane l: col N=l&15, holds K = k0+2*(l>>4)+{0,1}
// f32 16x16 C layout : vgpr r, lane l: M = r + 8*(l>>4), N = l&15

template <bool FUSE, int H>
__global__ __launch_bounds__(256)
void k_gemm(const float* __restrict__ A, const float* __restrict__ W,
            const float* __restrict__ biasIn, float* __restrict__ C,
            int N, int totalTiles) {
    constexpr int tilesN2 = ((H + 15) / 16 + 1) / 2;   // 16x32 tiles across H
    const int lane = threadIdx.x & 31;
    const int l15  = lane & 15;
    const int hi   = lane >> 4;
    const int wavesPerBlock = blockDim.x >> 5;
    const int wstride = gridDim.x * wavesPerBlock;
    int tile = blockIdx.x * wavesPerBlock + (threadIdx.x >> 5);

    for (; tile < totalTiles; tile += wstride) {
        const int tm = tile / tilesN2;
        const int tn = tile - tm * tilesN2;
        const int m0 = tm << 4;
        const int n0 = tn << 5;

        int mA = m0 + l15; if (mA > N - 1) mA = N - 1;        // clamp tail rows
        const float* arow = A + (size_t)mA * 128 + (hi << 1);

        const int nA = n0 + l15;
        const int nB = n0 + 16 + l15;
        const bool okA = nA < H;
        const bool okB = nB < H;
        const int ncA = okA ? nA : (H - 1);                    // clamp tail cols
        const int ncB = okB ? nB : (H - 1);
        const float* wA = W + (size_t)(hi << 1) * H + ncA;     // +k0*H = imm off
        const float* wB = W + (size_t)(hi << 1) * H + ncB;
        const float* bi = FUSE ? (biasIn + (hi << 1)) : nullptr;

        v8f acc0 = {0.f,0.f,0.f,0.f,0.f,0.f,0.f,0.f};
        v8f acc1 = {0.f,0.f,0.f,0.f,0.f,0.f,0.f,0.f};

        #pragma unroll
        for (int k0 = 0; k0 < 128; k0 += 4) {
            v2f a = *(const v2f*)(arow + k0);
            if (FUSE) {
                v2f bb = *(const v2f*)(bi + k0);
                a.x = fmaxf(a.x + bb.x, 0.f);
                a.y = fmaxf(a.y + bb.y, 0.f);
            }
            v2f b0, b1;
            b0.x = wA[k0 * H];
            b0.y = wA[(k0 + 1) * H];
            b1.x = wB[k0 * H];
            b1.y = wB[(k0 + 1) * H];

            acc0 = __builtin_amdgcn_wmma_f32_16x16x4_f32(
                       false, a, false, b0, (short)0, acc0, false, false);
            acc1 = __builtin_amdgcn_wmma_f32_16x16x4_f32(
                       false, a, false, b1, (short)0, acc1, false, false);
        }

        #pragma unroll
        for (int r = 0; r < 8; ++r) {
            const int m = m0 + r + (hi << 3);
            if (m < N) {
                if (okA) C[(size_t)m * H + nA] = acc0[r];
                if (okB) C[(size_t)m * H + nB] = acc1[r];
            }
        }
    }
}

// Runtime-H fallback (only used if the classifier width is not 40).
template <bool FUSE>
__global__ __launch_bounds__(256)
void k_gemm_rt(const float* __restrict__ A, const float* __restrict__ W,
               const float* __restrict__ biasIn, float* __restrict__ C,
               int N, int H, int tilesN2, int totalTiles) {
    const int lane = threadIdx.x & 31;
    const int l15  = lane & 15;
    const int hi   = lane >> 4;
    const int wavesPerBlock = blockDim.x >> 5;
    const int wstride = gridDim.x * wavesPerBlock;
    int tile = blockIdx.x * wavesPerBlock + (threadIdx.x >> 5);

    for (; tile < totalTiles; tile += wstride) {
        const int tm = tile / tilesN2;
        const int tn = tile - tm * tilesN2;
        const int m0 = tm << 4;
        const int n0 = tn << 5;

        int mA = m0 + l15; if (mA > N - 1) mA = N - 1;
        const float* arow = A + (size_t)mA * 128 + (hi << 1);

        const int nA = n0 + l15;
        const int nB = n0 + 16 + l15;
        const bool okA = nA < H;
        const bool okB = nB < H;
        const int ncA = okA ? nA : (H - 1);
        const int ncB = okB ? nB : (H - 1);
        const float* wA = W + (size_t)(hi << 1) * H + ncA;
        const float* wB = W + (size_t)(hi << 1) * H + ncB;
        const float* bi = FUSE ? (biasIn + (hi << 1)) : nullptr;

        v8f acc0 = {0.f,0.f,0.f,0.f,0.f,0.f,0.f,0.f};
        v8f acc1 = {0.f,0.f,0.f,0.f,0.f,0.f,0.f,0.f};

        #pragma unroll 4
        for (int k0 = 0; k0 < 128; k0 += 4) {
            v2f a = *(const v2f*)(arow + k0);
            if (FUSE) {
                v2f bb = *(const v2f*)(bi + k0);
                a.x = fmaxf(a.x + bb.x, 0.f);
                a.y = fmaxf(a.y + bb.y, 0.f);
            }
            v2f b0, b1;
            b0.x = wA[(size_t)k0 * H];
            b0.y = wA[(size_t)(k0 + 1) * H];
            b1.x = wB[(size_t)k0 * H];
            b1.y = wB[(size_t)(k0 + 1) * H];

            acc0 = __builtin_amdgcn_wmma_f32_16x16x4_f32(
                       false, a, false, b0, (short)0, acc0, false, false);
            acc1 = __builtin_amdgcn_wmma_f32_16x16x4_f32(
                       false, a, false, b1, (short)0, acc1, false, false);
        }

        #pragma unroll
        for (int r = 0; r < 8; ++r) {
            const int m = m0 + r + (hi << 3);
            if (m < N) {
                if (okA) C[(size_t)m * H + nA] = acc0[r];
                if (okB) C[(size_t)m * H + nB] = acc1[r];
            }
        }
    }
}

// ---------------------------- edge aggregation -----------------------------
// agg[d, f] += t[s, f] * norm[e].  Each thread handles 4 consecutive floats
// (float4 gather + 4 fp32 HW atomics); F/4 consecutive threads per edge.

template <int F>  // F % 4 == 0
__global__ void k_scatter4(const float* __restrict__ t, const int* __restrict__ s32,
                           const int* __restrict__ d32, const float* __restrict__ nrm,
                           float* __restrict__ agg, long long total4) {
    constexpr int Q = F / 4;
    long long idx = (long long)blockIdx.x * blockDim.x + threadIdx.x;
    const long long stride = (long long)gridDim.x * blockDim.x;
    for (; idx < total4; idx += stride) {
        const int e  = (int)(idx / Q);
        const int f4 = (int)(idx - (long long)e * Q) << 2;
        const int s = s32[e], d = d32[e];
        const float w = nrm[e];
        v4f v = *(const v4f*)(t + (size_t)s * F + f4);
        float* out = agg + (size_t)d * F + f4;
        atomAddF(out + 0, v.x * w);
        atomAddF(out + 1, v.y * w);
        atomAddF(out + 2, v.z * w);
        atomAddF(out + 3, v.w * w);
    }
}

__global__ void k_scatter_rt(const float* __restrict__ t, const int* __restrict__ s32,
                             const int* __restrict__ d32, const float* __restrict__ nrm,
                             float* __restrict__ agg, long long total, int F) {
    long long idx = (long long)blockIdx.x * blockDim.x + threadIdx.x;
    const long long stride = (long long)gridDim.x * blockDim.x;
    for (; idx < total; idx += stride) {
        const int e = (int)(idx / F);
        const int f = (int)(idx - (long long)e * F);
        const int s = s32[e], d = d32[e];
        atomAddF(&agg[(size_t)d * F + f], t[(size_t)s * F + f] * nrm[e]);
    }
}

// --------------------------- bias + log_softmax ----------------------------
// One wave per node; C <= 64 (reference: C = 40). wave32 shuffle reductions.

__global__ void k_logsoftmax(const float* __restrict__ agg, const float* __restrict__ b,
                             float* __restrict__ out, int N, int C) {
    const int node = blockIdx.x * (blockDim.x >> 5) + (threadIdx.x >> 5);
    const int lane = threadIdx.x & 31;
    if (node >= N) return;
    const float* row = agg + (size_t)node * C;
    const float NEG_INF = -__builtin_inff();
    float x0 = (lane < C)      ? row[lane]      + b[lane]      : NEG_INF;
    float x1 = (lane + 32 < C) ? row[lane + 32] + b[lane + 32] : NEG_INF;

    float m = fmaxf(x0, x1);
    for (int off = 16; off > 0; off >>= 1) m = fmaxf(m, __shfl_xor(m, off, 32));

    float s = expf(x0 - m) + ((lane + 32 < C) ? expf(x1 - m) : 0.f);
    for (int off = 16; off > 0; off >>= 1) s += __shfl_xor(s, off, 32);

    const float lse = logf(s) + m;
    if (lane < C)      out[(size_t)node * C + lane]      = x0 - lse;
    if (lane + 32 < C) out[(size_t)node * C + lane + 32] = x1 - lse;
}

// ------------------------------- launcher ----------------------------------

extern "C" void kernel_launch(void* const* d_in, const int* in_sizes, int n_in,
                              void* d_out, int out_size, void* d_ws, size_t ws_size,
                              hipStream_t stream) {
    const float* x  = (const float*)d_in[0];
    const int*   ei = (const int*)d_in[1];        // edge_index [2, E] (int32)
    const float* W1 = (const float*)d_in[2];
    const float* b1 = (const float*)d_in[3];
    const float* W2 = (const float*)d_in[4];
    const float* b2 = (const float*)d_in[5];
    const float* W3 = (const float*)d_in[6];
    const float* b3 = (const float*)d_in[7];

    const int FDIM = 128, H = 128;
    const int N = in_sizes[0] / FDIM;
    const int E = in_sizes[1] / 2;
    const int C = in_sizes[7];
    const long long EN = (long long)E + N;

    // workspace carve-out
    char* w = (char*)d_ws;
    auto take = [&](size_t bytes) -> char* {
        char* p = w; w += (bytes + 255) & ~(size_t)255; return p;
    };
    float* bufT   = (float*)take((size_t)N * FDIM * sizeof(float));
    float* bufAgg = (float*)take((size_t)N * FDIM * sizeof(float));
    float* dinv   = (float*)take((size_t)N * sizeof(float));
    float* nrm    = (float*)take((size_t)EN * sizeof(float));
    int*   s32    = (int*)take((size_t)EN * sizeof(int));
    int*   d32    = (int*)take((size_t)EN * sizeof(int));

    const int* esrc = ei;
    const int* edst = ei + E;

    const int T = 256;
    const int tilesM  = (N + 15) / 16;
    const int tilesN2 = ((H + 15) / 16 + 1) / 2;            // 16x32 tiles
    const int tilesC2 = ((C + 15) / 16 + 1) / 2;
    auto blk  = [](long long n, int t) { return dim3((unsigned)((n + t - 1) / t)); };
    auto gblk = [](long long n) {                            // capped grid-stride
        long long b = (n + 255) / 256; if (b > (1 << 20)) b = 1 << 20;
        return dim3((unsigned)b);
    };

    // --- normalization: deg = 1 (self loop) + in-degree; dinv = rsqrt(deg) ---
    k_fill  <<<blk(N, T), T, 0, stream>>>(dinv, 1.0f, (long long)N);
    k_deg   <<<blk(E, T), T, 0, stream>>>(edst, dinv, E);
    k_rsqrt <<<blk(N, T), T, 0, stream>>>(dinv, N);
    k_edges <<<blk(EN, T), T, 0, stream>>>(esrc, edst, dinv, s32, d32, nrm, E, N);

    const long long aggElems = (long long)N * FDIM;
    const long long scat128q = EN * (FDIM / 4);

    // --- layer 1: t = x @ W1 ; agg = scatter(t) ---
    {
        int total = tilesM * tilesN2;
        k_gemm<false, 128><<<blk((long long)total, 8), T, 0, stream>>>(x, W1, nullptr, bufT, N, total);
    }
    k_fill<<<gblk(aggElems), T, 0, stream>>>(bufAgg, 0.f, aggElems);
    k_scatter4<128><<<gblk(scat128q), T, 0, stream>>>(bufT, s32, d32, nrm, bufAgg, scat128q);

    // --- layer 2: t = relu(agg + b1) @ W2 ; agg = scatter(t) ---
    {
        int total = tilesM * tilesN2;
        k_gemm<true, 128><<<blk((long long)total, 8), T, 0, stream>>>(bufAgg, W2, b1, bufT, N, total);
    }
    k_fill<<<gblk(aggElems), T, 0, stream>>>(bufAgg, 0.f, aggElems);
    k_scatter4<128><<<gblk(scat128q), T, 0, stream>>>(bufT, s32, d32, nrm, bufAgg, scat128q);

    // --- layer 3: t = relu(agg + b2) @ W3 ; agg = scatter(t) ; logsoftmax ---
    if (C == 40) {
        int total = tilesM * ((40 / 16 + 1 + 1) / 2);        // tilesC2 for H=40 -> 2
        k_gemm<true, 40><<<blk((long long)total, 8), T, 0, stream>>>(bufAgg, W3, b2, bufT, N, total);
    } else {
        int total = tilesM * tilesC2;
        k_gemm_rt<true><<<blk((long long)total, 8), T, 0, stream>>>(bufAgg, W3, b2, bufT, N, C, tilesC2, total);
    }
    const long long aggC = (long long)N * C;
    k_fill<<<gblk(aggC), T, 0, stream>>>(bufAgg, 0.f, aggC);
    if (C == 40) {
        const long long scatCq = EN * (40 / 4);
        k_scatter4<40><<<gblk(scatCq), T, 0, stream>>>(bufT, s32, d32, nrm, bufAgg, scatCq);
    } else {
        const long long scatC = EN * C;
        k_scatter_rt<<<gblk(scatC), T, 0, stream>>>(bufT, s32, d32, nrm, bufAgg, scatC, C);
    }

    k_logsoftmax<<<blk(N, 8), T, 0, stream>>>(bufAgg, b3, (float*)d_out, N, C);
}